// MyAvgPool_15315853378255
// MI455X (gfx1250) — compile-verified
//
#include <hip/hip_runtime.h>

// MyAvgPool: 2x2 avg pool, stride 2, pad 0.
// Input  (N,C,H,W) = (32,192,112,112) fp32  -> 77,070,336 elems
// Output (N,C,56,56)                  fp32  -> 19,267,584 elems
//
// Memory-bound (385 MB total traffic, ~16.5us floor at 23.3 TB/s).
// The 2x2 window reduction is performed with V_WMMA_F32_16X16X4_F32:
//   A = all 0.25 (16x4)  =>  D[m][n] = 0.25 * sum_k B[k][n]  for all m.
// Column n of B receives the 4 window values of output n (k-order
// irrelevant by permutation invariance), and the pooled value is
// replicated over all m, so any D slot of column n is the answer.

typedef __attribute__((ext_vector_type(2))) float v2f;
typedef __attribute__((ext_vector_type(8))) float v8f;

__global__ __launch_bounds__(256) void avgpool2x2_wmma_kernel(
    const float* __restrict__ in, float* __restrict__ out)
{
    const unsigned tid   = threadIdx.x;
    const unsigned lane  = tid & 31u;
    const unsigned l15   = lane & 15u;
    const unsigned rsel  = lane >> 4;          // 0 = top row of window, 1 = bottom

    // wave id within the whole grid (8 waves per 256-thread block)
    const unsigned waveId = blockIdx.x * 8u + (tid >> 5);
    const unsigned o0     = waveId * 32u;      // first of 32 outputs for this wave

    // ---- group A: outputs o0 .. o0+15 -------------------------------------
    // flattened output index -> q = plane*56 + h_out, w = w_out
    // input float2 offset = 224*q + rsel*112 + 2*w   (W=112, 2 rows/window)
    const unsigned oa = o0 + l15;
    const unsigned qa = oa / 56u;
    const unsigned wa = oa - qa * 56u;
    const unsigned offA = qa * 224u + rsel * 112u + 2u * wa;

    // ---- group B: outputs o0+16 .. o0+31 ----------------------------------
    const unsigned ob = oa + 16u;
    const unsigned qb = ob / 56u;
    const unsigned wb = ob - qb * 56u;
    const unsigned offB = qb * 224u + rsel * 112u + 2u * wb;

    // Each half-wave loads a contiguous 128B run (fully coalesced B64 loads).
    const v2f bA = *(const v2f*)(in + offA);
    const v2f bB = *(const v2f*)(in + offB);

    const v2f a = {0.25f, 0.25f};   // all-0.25 A matrix (layout invariant)
    const v8f c = {};               // zero accumulator

    // D1[m][n] = pooled value of output (o0 + n), replicated over m
    v8f d1 = __builtin_amdgcn_wmma_f32_16x16x4_f32(
        /*neg_a=*/false, a, /*neg_b=*/false, bA,
        /*c_mod=*/(short)0, c, /*reuse_a=*/false, /*reuse_b=*/false);
    // D2[m][n] = pooled value of output (o0 + 16 + n), replicated over m
    v8f d2 = __builtin_amdgcn_wmma_f32_16x16x4_f32(
        /*neg_a=*/false, a, /*neg_b=*/false, bB,
        /*c_mod=*/(short)0, c, /*reuse_a=*/false, /*reuse_b=*/false);

    // Lane L (0..15) holds pooled(o0+L) in d1[0]; lane L (16..31) holds
    // pooled(o0+L) in d2[0] (column L-16 of D2 lives in lanes 16..31 too).
    const float val = (rsel == 0u) ? d1[0] : d2[0];
    out[o0 + lane] = val;   // 32 consecutive dwords -> 128B coalesced store
}

extern "C" void kernel_launch(void* const* d_in, const int* in_sizes, int n_in,
                              void* d_out, int out_size, void* d_ws, size_t ws_size,
                              hipStream_t stream)
{
    (void)in_sizes; (void)n_in; (void)d_ws; (void)ws_size;
    const float* in  = (const float*)d_in[0];
    float*       out = (float*)d_out;

    // out_size = 19,267,584 = 75,264 * 256 exactly: every wave is full,
    // EXEC is all-1s (required for WMMA), no tail handling needed.
    const int threads = 256;
    const int blocks  = out_size / threads;   // 75,264
    avgpool2x2_wmma_kernel<<<blocks, threads, 0, stream>>>(in, out);
}